// TnoCausalV11_69621419869056
// MI455X (gfx1250) — compile-verified
//
#include <hip/hip_runtime.h>
#include <math.h>
#include <stdint.h>

// ---------------------------------------------------------------------------
// Causal per-channel Toeplitz conv via four-step FFT (N = 16384 = 128 x 128),
// all 128-point DFT stages as fp32 WMMA complex GEMMs (V_WMMA_F32_16X16X4_F32).
// One workgroup (256 thr / 8 waves) owns one (b,d) channel; whole 16384-pt
// complex signal resident in LDS (2 ping-pong complex planes, ~264 KB of the
// 320 KB WGP LDS). DFT matrix is prepacked into per-lane WMMA fragment order
// (L2-resident, coalesced 8B/lane loads). Signal rows enter LDS via the
// Tensor Data Mover with hardware padding to the conflict-free LDS stride.
// ---------------------------------------------------------------------------

typedef __attribute__((ext_vector_type(2))) float v2f;
typedef __attribute__((ext_vector_type(8))) float v8f;
typedef unsigned int  u32x4 __attribute__((ext_vector_type(4)));
typedef int           i32x4 __attribute__((ext_vector_type(4)));
typedef int           i32x8 __attribute__((ext_vector_type(8)));

#define NSEQ  8192
#define NFFT  16384
#define RADIX 128
#define NCH   512
#define NB    4
#define LSTR  132                 // LDS row stride (floats): 128 + 4 pad
#define PLANE (RADIX * LSTR)      // 16896 floats / plane

// workspace layout (float offsets)
#define WS_TWRE ((size_t)0)                      // twiddle exp(-2pi i a c/N)
#define WS_TWIM (WS_TWRE + RADIX * RADIX)
#define WS_WARE (WS_TWIM + RADIX * RADIX)        // W in A-fragment order
#define WS_WAIM (WS_WARE + RADIX * RADIX)
#define WS_WBRE (WS_WAIM + RADIX * RADIX)        // W in B-fragment order
#define WS_WBIM (WS_WBRE + RADIX * RADIX)
#define WS_TF   (WS_WBIM + RADIX * RADIX)        // 512 ch * 2*NFFT
#define WS_XTT  (WS_TF + (size_t)NCH * 2 * NFFT) // 2560 rows * NSEQ

// ---------------------------------------------------------------------------
// Twiddle table TW[a][c] = exp(-2*pi*i*a*c/16384)
// ---------------------------------------------------------------------------
__global__ void tno_build_tables(float* ws) {
  int a = blockIdx.x, c = threadIdx.x;
  const double PI2 = 6.283185307179586476925286766559;
  double ang = -PI2 * (double)(a * c) / (double)NFFT;
  ws[WS_TWRE + a * RADIX + c] = (float)cos(ang);
  ws[WS_TWIM + a * RADIX + c] = (float)sin(ang);
}

// ---------------------------------------------------------------------------
// Prepack DFT-128 matrix W[r][c]=exp(-2pi i rc/128) into WMMA per-lane order.
// A-layout [blk][ks][lane]{2}: row=blk*16+lane%16, k=ks*4+(lane/16)*2 +{0,1}
// B-layout [blk][ks][lane]{2}: col=blk*16+lane%16, k=ks*4+(lane/16) +{0,2}
// ---------------------------------------------------------------------------
__global__ void tno_build_packs(float* ws) {
  int idx = blockIdx.x * 256 + threadIdx.x;     // 0..8191 = [blk8][ks32][lane32]
  int lane = idx & 31, ks = (idx >> 5) & 31, blk = idx >> 10;
  int m15 = lane & 15, hi = lane >> 4;
  const double PI2 = 6.283185307179586476925286766559;
  int rowA = blk * 16 + m15, kA = ks * 4 + hi * 2;
  double a0 = -PI2 * (double)(rowA * kA) / 128.0;
  double a1 = -PI2 * (double)(rowA * (kA + 1)) / 128.0;
  ws[WS_WARE + idx * 2]     = (float)cos(a0);
  ws[WS_WARE + idx * 2 + 1] = (float)cos(a1);
  ws[WS_WAIM + idx * 2]     = (float)sin(a0);
  ws[WS_WAIM + idx * 2 + 1] = (float)sin(a1);
  int colB = blk * 16 + m15, kB = ks * 4 + hi;
  double b0 = -PI2 * (double)(kB * colB) / 128.0;
  double b1 = -PI2 * (double)((kB + 2) * colB) / 128.0;
  ws[WS_WBRE + idx * 2]     = (float)cos(b0);
  ws[WS_WBRE + idx * 2 + 1] = (float)cos(b1);
  ws[WS_WBIM + idx * 2]     = (float)sin(b0);
  ws[WS_WBIM + idx * 2 + 1] = (float)sin(b1);
}

// ---------------------------------------------------------------------------
// Coalesced layout shufflers: (b,n,d)/(n,d) <-> (b*D+d, n) channel rows
// ---------------------------------------------------------------------------
__global__ void tno_transpose_in(const float* __restrict__ x,
                                 const float* __restrict__ t,
                                 float* __restrict__ xtt) {
  __shared__ float tile[32][33];
  int n0 = blockIdx.x * 32, d0 = blockIdx.y * 32, z = blockIdx.z;
  const float* src = (z < NB) ? (x + (size_t)z * NSEQ * NCH) : t;
  for (int i = threadIdx.y; i < 32; i += 8)
    tile[i][threadIdx.x] = src[(size_t)(n0 + i) * NCH + d0 + threadIdx.x];
  __syncthreads();
  for (int i = threadIdx.y; i < 32; i += 8)
    xtt[(size_t)(z * NCH + d0 + i) * NSEQ + n0 + threadIdx.x] = tile[threadIdx.x][i];
}

__global__ void tno_transpose_out(const float* __restrict__ xtt,
                                  float* __restrict__ out) {
  __shared__ float tile[32][33];
  int n0 = blockIdx.x * 32, d0 = blockIdx.y * 32, b = blockIdx.z;
  for (int i = threadIdx.y; i < 32; i += 8)
    tile[i][threadIdx.x] = xtt[(size_t)(b * NCH + d0 + i) * NSEQ + n0 + threadIdx.x];
  __syncthreads();
  for (int i = threadIdx.y; i < 32; i += 8)
    out[((size_t)b * NSEQ + n0 + i) * NCH + d0 + threadIdx.x] = tile[threadIdx.x][i];
}

// ---------------------------------------------------------------------------
// TDM: contiguous 8192-dword global row -> LDS, with HW padding of 4 dwords
// every 128 dwords (produces the LSTR=132 bank-conflict-free layout).
// D# per CDNA5 ISA. Issued by one wave, EXEC ignored.
// ---------------------------------------------------------------------------
__device__ __attribute__((always_inline))
static void tdm_load_row_to_lds(const float* gsrc, unsigned lds_byte_off) {
  unsigned long long ga = (unsigned long long)(uintptr_t)gsrc;
  u32x4 g0;
  g0[0] = 1u;                                        // count=1 (valid user D#)
  g0[1] = lds_byte_off;                              // lds_addr
  g0[2] = (unsigned)(ga & 0xFFFFFFFFu);              // global_addr[31:0]
  g0[3] = (unsigned)((ga >> 32) & 0x01FFFFFFu) | (2u << 30);  // [56:32] | type=2
  i32x8 g1;
  // data_size=2 (4B) | pad_enable | pad_interval=6 (128dw) | pad_amount=3 (4dw)
  g1[0] = (2 << 16) | (1 << 20) | (6 << 22) | (3 << 25);
  g1[1] = (NSEQ & 0xFFFF) << 16;                     // tensor_dim0 = 8192 (lo16)
  g1[2] = (1 << 16);                                 // tensor_dim0 hi=0 | tensor_dim1=1
  g1[3] = (NSEQ & 0xFFFF) << 16;                     // tile_dim0 = 8192
  g1[4] = 1;                                         // tile_dim1 = 1, tile_dim2 = 0
  g1[5] = NSEQ;                                      // tensor_dim0_stride lo32
  g1[6] = 0;                                         // stride hi | dim1_stride lo
  g1[7] = 0;
  i32x4 gz = {0, 0, 0, 0};
#if defined(__clang_major__) && (__clang_major__ >= 23)
  i32x8 gz8 = {0, 0, 0, 0, 0, 0, 0, 0};
  __builtin_amdgcn_tensor_load_to_lds(g0, g1, gz, gz, gz8, 0);
#else
  __builtin_amdgcn_tensor_load_to_lds(g0, g1, gz, gz, 0);
#endif
}

// ---------------------------------------------------------------------------
// 128x128x128 complex GEMM on V_WMMA_F32_16X16X4_F32.
//  - Diagonal tile walk (tm=(wave+i)&7, tn=i): no loop-invariant fragments,
//    so nothing gets hoisted/spilled.
//  - Conjugation folded via imag-plane sign; 2 accumulator chains per tile.
//  - PACK operands: coalesced global loads of the prepacked W fragments.
// ---------------------------------------------------------------------------
template<bool A_PACK, bool B_PACK>
__device__ __attribute__((always_inline))
static void cgemm128(const float* __restrict__ aRe, const float* __restrict__ aIm,
                     float aSign,
                     const float* __restrict__ bRe, const float* __restrict__ bIm,
                     float bSign,
                     float* __restrict__ oRe, float* __restrict__ oIm) {
  int lane = threadIdx.x & 31, w = threadIdx.x >> 5;
  int m15 = lane & 15, hi = lane >> 4;
#pragma unroll 1
  for (int i = 0; i < 8; ++i) {
    int tm = (w + i) & 7, tn = i;
    v8f cre = {}, cim = {};
#pragma unroll 4
    for (int ks = 0; ks < 32; ++ks) {
      v2f ar, ai0, br, bi0;
      if (A_PACK) {
        int off = ((tm * 32 + ks) * 32 + lane) * 2;
        ar  = { aRe[off], aRe[off + 1] };
        ai0 = { aIm[off], aIm[off + 1] };
      } else {
        int ra = (tm * 16 + m15) * LSTR + ks * 4 + hi * 2;
        ar  = { aRe[ra], aRe[ra + 1] };
        ai0 = { aIm[ra], aIm[ra + 1] };
      }
      if (B_PACK) {
        int off = ((tn * 32 + ks) * 32 + lane) * 2;
        br  = { bRe[off], bRe[off + 1] };
        bi0 = { bIm[off], bIm[off + 1] };
      } else {
        int r0 = (ks * 4 + hi) * LSTR + tn * 16 + m15;
        br  = { bRe[r0], bRe[r0 + 2 * LSTR] };
        bi0 = { bIm[r0], bIm[r0 + 2 * LSTR] };
      }
      v2f ai  = { aSign * ai0.x, aSign * ai0.y };
      v2f bi  = { bSign * bi0.x, bSign * bi0.y };
      v2f ain = { -ai.x, -ai.y };
      cre = __builtin_amdgcn_wmma_f32_16x16x4_f32(false, ar,  false, br, (short)0, cre, false, false);
      cre = __builtin_amdgcn_wmma_f32_16x16x4_f32(false, ain, false, bi, (short)0, cre, false, false);
      cim = __builtin_amdgcn_wmma_f32_16x16x4_f32(false, ai,  false, br, (short)0, cim, false, false);
      cim = __builtin_amdgcn_wmma_f32_16x16x4_f32(false, ar,  false, bi, (short)0, cim, false, false);
    }
#pragma unroll
    for (int v = 0; v < 8; ++v) {
      int row = tm * 16 + v + 8 * hi;
      int col = tn * 16 + m15;
      oRe[row * LSTR + col] = cre[v];
      oIm[row * LSTR + col] = cim[v];
    }
  }
}

// Forward four-step: S[k2][k1] = (TW ⊙ (W·X)) · W
__device__ __attribute__((always_inline))
static void fft_forward(float* Ure, float* Uim, float* Vre, float* Vim,
                        const float* ws) {
  __builtin_prefetch(ws + WS_WARE, 0, 0);
  __builtin_prefetch(ws + WS_WBRE, 0, 0);
  cgemm128<true, false>(ws + WS_WARE, ws + WS_WAIM, 1.0f, Ure, Uim, 1.0f, Vre, Vim);
  __syncthreads();
  const float* Tre = ws + WS_TWRE; const float* Tim = ws + WS_TWIM;
  for (int e = threadIdx.x; e < RADIX * RADIX; e += 256) {
    int r = e >> 7, c = e & 127;
    float vr = Vre[r * LSTR + c], vi = Vim[r * LSTR + c];
    float tr = Tre[e], ti = Tim[e];
    Vre[r * LSTR + c] = vr * tr - vi * ti;
    Vim[r * LSTR + c] = vr * ti + vi * tr;
  }
  __syncthreads();
  cgemm128<false, true>(Vre, Vim, 1.0f, ws + WS_WBRE, ws + WS_WBIM, 1.0f, Ure, Uim);
  __syncthreads();
}

// Inverse (1/N applied at store): Y = conj(W) · (conj(TW) ⊙ (S · conj(W)))
__device__ __attribute__((always_inline))
static void fft_inverse(float* Ure, float* Uim, float* Vre, float* Vim,
                        const float* ws) {
  cgemm128<false, true>(Ure, Uim, 1.0f, ws + WS_WBRE, ws + WS_WBIM, -1.0f, Vre, Vim);
  __syncthreads();
  const float* Tre = ws + WS_TWRE; const float* Tim = ws + WS_TWIM;
  for (int e = threadIdx.x; e < RADIX * RADIX; e += 256) {
    int r = e >> 7, c = e & 127;
    float vr = Vre[r * LSTR + c], vi = Vim[r * LSTR + c];
    float tr = Tre[e], ti = -Tim[e];
    Vre[r * LSTR + c] = vr * tr - vi * ti;
    Vim[r * LSTR + c] = vr * ti + vi * tr;
  }
  __syncthreads();
  cgemm128<true, false>(ws + WS_WARE, ws + WS_WAIM, -1.0f, Vre, Vim, 1.0f, Ure, Uim);
  __syncthreads();
}

__device__ __attribute__((always_inline))
static float gelu_tanh(float v) {
  const float k0 = 0.7978845608028654f, k1 = 0.044715f;
  float u = k0 * (v + k1 * v * v * v);
  return 0.5f * v * (1.0f + tanhf(u));
}

// Shared loader: TDM for the 8192-dword row (wave 0), threads zero the rest.
__device__ __attribute__((always_inline))
static void load_row_to_planes(const float* src, float* Ure, float* Uim) {
  if ((threadIdx.x >> 5) == 0) {
    tdm_load_row_to_lds(src, 0u);            // U_re lives at dyn-LDS offset 0
  }
  for (int idx = threadIdx.x; idx < NFFT; idx += 256) {
    int r = idx >> 7, c = idx & 127;
    if (idx >= NSEQ) Ure[r * LSTR + c] = 0.0f;
    Uim[r * LSTR + c] = 0.0f;
  }
  if ((threadIdx.x >> 5) == 0) {
    __builtin_amdgcn_s_wait_tensorcnt((short)0);
  }
  __syncthreads();
}

// ---------------------------------------------------------------------------
// Filter spectrum: one WG per channel d. Tf[d] = FFT(pad(t[:,d]))
// ---------------------------------------------------------------------------
__global__ void __launch_bounds__(256) tno_fft_filter(float* ws) {
  extern __shared__ float lds[];
  float* Ure = lds;             float* Uim = lds + PLANE;
  float* Vre = lds + 2 * PLANE; float* Vim = lds + 3 * PLANE;
  int d = blockIdx.x;
  load_row_to_planes(ws + WS_XTT + (size_t)(NB * NCH + d) * NSEQ, Ure, Uim);
  fft_forward(Ure, Uim, Vre, Vim, ws);
  float* tf = ws + WS_TF + (size_t)d * (2 * NFFT);
  for (int idx = threadIdx.x; idx < NFFT; idx += 256) {
    int r = idx >> 7, c = idx & 127;
    tf[idx]        = Ure[r * LSTR + c];
    tf[NFFT + idx] = Uim[r * LSTR + c];
  }
}

// ---------------------------------------------------------------------------
// Fused conv: one WG per (b,d) row. FFT -> ⊙Tf[d] -> IFFT -> gelu -> row.
// ---------------------------------------------------------------------------
__global__ void __launch_bounds__(256) tno_conv_fft(float* ws) {
  extern __shared__ float lds[];
  float* Ure = lds;             float* Uim = lds + PLANE;
  float* Vre = lds + 2 * PLANE; float* Vim = lds + 3 * PLANE;
  int row = blockIdx.x;         // b*NCH + d
  int d = row & (NCH - 1);
  float* xr = ws + WS_XTT + (size_t)row * NSEQ;
  const float* tf = ws + WS_TF + (size_t)d * (2 * NFFT);
  __builtin_prefetch(tf, 0, 0);
  load_row_to_planes(xr, Ure, Uim);
  fft_forward(Ure, Uim, Vre, Vim, ws);
  for (int idx = threadIdx.x; idx < NFFT; idx += 256) {
    int r = idx >> 7, c = idx & 127;
    float sr = Ure[r * LSTR + c], si = Uim[r * LSTR + c];
    float fr = tf[idx], fi = tf[NFFT + idx];
    Ure[r * LSTR + c] = sr * fr - si * fi;
    Uim[r * LSTR + c] = sr * fi + si * fr;
  }
  __syncthreads();
  fft_inverse(Ure, Uim, Vre, Vim, ws);
  const float invN = 1.0f / (float)NFFT;
  for (int idx = threadIdx.x; idx < NSEQ; idx += 256) {
    int r = idx >> 7, c = idx & 127;
    xr[idx] = gelu_tanh(Ure[r * LSTR + c] * invN);
  }
}

// ---------------------------------------------------------------------------
extern "C" void kernel_launch(void* const* d_in, const int* in_sizes, int n_in,
                              void* d_out, int out_size, void* d_ws, size_t ws_size,
                              hipStream_t stream) {
  (void)in_sizes; (void)n_in; (void)out_size; (void)ws_size;
  const float* x = (const float*)d_in[0];
  const float* t = (const float*)d_in[1];
  float* out = (float*)d_out;
  float* ws  = (float*)d_ws;

  const size_t ldsBytes = (size_t)4 * PLANE * sizeof(float);  // ~264 KB (<=320 KB WGP LDS)
  hipFuncSetAttribute(reinterpret_cast<const void*>(tno_fft_filter),
                      hipFuncAttributeMaxDynamicSharedMemorySize, (int)ldsBytes);
  hipFuncSetAttribute(reinterpret_cast<const void*>(tno_conv_fft),
                      hipFuncAttributeMaxDynamicSharedMemorySize, (int)ldsBytes);

  tno_build_tables<<<dim3(RADIX), dim3(RADIX), 0, stream>>>(ws);
  tno_build_packs<<<dim3(32), dim3(256), 0, stream>>>(ws);
  tno_transpose_in<<<dim3(NSEQ / 32, NCH / 32, NB + 1), dim3(32, 8), 0, stream>>>(x, t, ws + WS_XTT);
  tno_fft_filter<<<dim3(NCH), dim3(256), ldsBytes, stream>>>(ws);
  tno_conv_fft<<<dim3(NB * NCH), dim3(256), ldsBytes, stream>>>(ws);
  tno_transpose_out<<<dim3(NSEQ / 32, NCH / 32, NB), dim3(32, 8), 0, stream>>>(ws + WS_XTT, out);
}